// RankPooling_69827578298756
// MI455X (gfx1250) — compile-verified
//
#include <hip/hip_runtime.h>
#include <math.h>

typedef float v2f __attribute__((ext_vector_type(2)));
typedef float v8f __attribute__((ext_vector_type(8)));

constexpr int NB = 64;
constexpr int ND = 512;
constexpr int NT = 512;
constexpr int N_ITERS = 500;
constexpr float C_SVR   = 1.0f;
constexpr float EPS_SVR = 0.1f;
constexpr float LR0     = 0.1f;
constexpr float DECAY   = 1e-3f;

constexpr int KL_ROWS   = 128;                       // K rows cached in LDS
constexpr size_t KL_BYTES = (size_t)KL_ROWS * NT * sizeof(float);   // 256 KB

// per-wave row split in the matvec: 8 LDS rows + 24 global rows per wave
constexpr int ROWS_LDS_PER_WAVE  = KL_ROWS / 16;        // 8
constexpr int ROWS_GLB_PER_WAVE  = (NT - KL_ROWS) / 16; // 24

// ---------------------------------------------------------------------------
// Kernel A: cumulative mean over T, written transposed into ws X[B, T, D].
// ---------------------------------------------------------------------------
__global__ void k_cummean(const float* __restrict__ in, float* __restrict__ X) {
    int r = blockIdx.x * blockDim.x + threadIdx.x;   // r = b*ND + d
    int b = r >> 9;
    int d = r & (ND - 1);
    const float* src = in + (size_t)r * NT;          // input [B, D, T]
    float* dst = X + (size_t)b * NT * ND + d;        // X[b, t, d]
    float s = 0.0f;
    for (int t = 0; t < NT; ++t) {
        s += src[t];
        dst[(size_t)t * ND] = s / (float)(t + 1);
    }
}

// ---------------------------------------------------------------------------
// Kernel B: per-(b,t) L2 normalization over D, in place.
// ---------------------------------------------------------------------------
__global__ void k_norm(float* __restrict__ X) {
    __shared__ float red[4];
    int tid = threadIdx.x;                            // 0..127
    float4* p = (float4*)(X + (size_t)blockIdx.x * ND) + tid;
    float4 v = *p;
    float s = v.x * v.x + v.y * v.y + v.z * v.z + v.w * v.w;
    #pragma unroll
    for (int off = 16; off > 0; off >>= 1) s += __shfl_xor(s, off, 32);
    if ((tid & 31) == 0) red[tid >> 5] = s;
    __syncthreads();
    float norm = sqrtf(red[0] + red[1] + red[2] + red[3]);
    v.x /= norm; v.y /= norm; v.z /= norm; v.w /= norm;
    *p = v;
}

// ---------------------------------------------------------------------------
// Kernel C: Gram K[b] = X[b] X[b]^T via V_WMMA_F32_16X16X4_F32.
// One wave computes a 32x32 macro-tile (2x2 WMMA tiles); each A/B fragment
// is reused by two WMMAs, halving global read traffic vs 1 tile/wave.
// ---------------------------------------------------------------------------
__global__ void k_gram(const float* __restrict__ X, float* __restrict__ K) {
    int wid  = (blockIdx.x * blockDim.x + threadIdx.x) >> 5;
    int lane = threadIdx.x & 31;
    int b    = wid >> 8;                   // 16*16 = 256 macro-tiles per batch
    int tile = wid & 255;
    int ti   = (tile >> 4) << 5;           // macro-tile row base
    int tj   = (tile & 15) << 5;           // macro-tile col base

    const float* Xb = X + (size_t)b * NT * ND;
    int koff = (lane >> 4) << 1;
    const float* A0 = Xb + (size_t)(ti +      (lane & 15)) * ND + koff;
    const float* A1 = Xb + (size_t)(ti + 16 + (lane & 15)) * ND + koff;
    const float* B0 = Xb + (size_t)(tj +      (lane & 15)) * ND + koff;
    const float* B1 = Xb + (size_t)(tj + 16 + (lane & 15)) * ND + koff;

    v8f acc00 = {}, acc01 = {}, acc10 = {}, acc11 = {};
    #pragma unroll 2
    for (int k = 0; k < ND; k += 4) {
        v2f a0 = *(const v2f*)(A0 + k);
        v2f a1 = *(const v2f*)(A1 + k);
        v2f b0 = *(const v2f*)(B0 + k);
        v2f b1 = *(const v2f*)(B1 + k);
        acc00 = __builtin_amdgcn_wmma_f32_16x16x4_f32(false, a0, false, b0, (short)0, acc00, false, false);
        acc01 = __builtin_amdgcn_wmma_f32_16x16x4_f32(false, a0, false, b1, (short)0, acc01, false, false);
        acc10 = __builtin_amdgcn_wmma_f32_16x16x4_f32(false, a1, false, b0, (short)0, acc10, false, false);
        acc11 = __builtin_amdgcn_wmma_f32_16x16x4_f32(false, a1, false, b1, (short)0, acc11, false, false);
    }

    // C/D layout: VGPR r, lanes 0-15 -> (M=r, N=lane); lanes 16-31 -> (M=8+r, N=lane-16)
    float* Kb = K + (size_t)b * NT * NT;
    int mo = (lane >> 4) << 3;
    int n  = lane & 15;
    #pragma unroll
    for (int r = 0; r < 8; ++r) {
        Kb[(size_t)(ti + mo + r) * NT + tj + n]            = acc00[r];
        Kb[(size_t)(ti + mo + r) * NT + tj + 16 + n]       = acc01[r];
        Kb[(size_t)(ti + 16 + mo + r) * NT + tj + n]       = acc10[r];
        Kb[(size_t)(ti + 16 + mo + r) * NT + tj + 16 + n]  = acc11[r];
    }
}

// ---------------------------------------------------------------------------
// Kernel D: dual-form SVR loop + final w = X^T a. One block (16 waves) per
// batch. First 128 K rows cached in 256 KB dynamic LDS; each wave handles a
// balanced mix of 8 LDS rows + 24 L2-streamed rows (all waves issue global
// loads -> better latency hiding, equal per-wave work). a/g/b live in LDS.
// ---------------------------------------------------------------------------
__global__ void k_svr(const float* __restrict__ Kmat, const float* __restrict__ X,
                      float* __restrict__ out) {
    extern __shared__ float Kl[];                    // KL_ROWS x NT
    __shared__ __align__(16) float aL[NT];
    __shared__ float gL[NT];
    __shared__ float red[16];
    __shared__ float bL;

    int tid  = threadIdx.x;          // 0..511
    int lane = tid & 31;
    int wave = tid >> 5;             // 0..15
    int b    = blockIdx.x;
    const float* Kb = Kmat + (size_t)b * NT * NT;

    // Stage first KL_ROWS rows of K into LDS (reused 500x).
    {
        const float4* Kg4 = (const float4*)Kb;
        float4* Kl4 = (float4*)Kl;
        for (int i = tid; i < (KL_ROWS * NT) / 4; i += 512) Kl4[i] = Kg4[i];
    }
    aL[tid] = 0.0f;
    if (tid == 0) bL = 0.0f;
    __syncthreads();

    const float4* aL4 = (const float4*)aL;
    for (int it = 0; it < N_ITERS; ++it) {
        float lr = LR0 / (1.0f + DECAY * (float)it);

        // cache this lane's slice of a (reused across this wave's rows)
        float4 a4[4];
        #pragma unroll
        for (int c = 0; c < 4; ++c) a4[c] = aL4[c * 32 + lane];

        // --- this wave's LDS-resident rows ---
        for (int rr = 0; rr < ROWS_LDS_PER_WAVE; ++rr) {
            int row = wave * ROWS_LDS_PER_WAVE + rr;
            const float4* Kr4 = (const float4*)(Kl + (size_t)row * NT);
            float s = 0.0f;
            #pragma unroll
            for (int c = 0; c < 4; ++c) {
                float4 k4 = Kr4[c * 32 + lane];
                s += k4.x * a4[c].x + k4.y * a4[c].y + k4.z * a4[c].z + k4.w * a4[c].w;
            }
            #pragma unroll
            for (int off = 16; off > 0; off >>= 1) s += __shfl_xor(s, off, 32);
            if (lane == 0) {
                float e = s + bL - (float)row;
                gL[row] = (e > EPS_SVR) ? 1.0f : ((e < -EPS_SVR) ? -1.0f : 0.0f);
            }
        }
        // --- this wave's L2-streamed rows (b128, 512B contiguous per load) ---
        for (int rr = 0; rr < ROWS_GLB_PER_WAVE; ++rr) {
            int row = KL_ROWS + wave * ROWS_GLB_PER_WAVE + rr;
            const float4* Kr4 = (const float4*)(Kb + (size_t)row * NT);
            float s = 0.0f;
            #pragma unroll
            for (int c = 0; c < 4; ++c) {
                float4 k4 = Kr4[c * 32 + lane];
                s += k4.x * a4[c].x + k4.y * a4[c].y + k4.z * a4[c].z + k4.w * a4[c].w;
            }
            #pragma unroll
            for (int off = 16; off > 0; off >>= 1) s += __shfl_xor(s, off, 32);
            if (lane == 0) {
                float e = s + bL - (float)row;
                gL[row] = (e > EPS_SVR) ? 1.0f : ((e < -EPS_SVR) ? -1.0f : 0.0f);
            }
        }
        __syncthreads();

        // sum(g) block reduction
        float g = gL[tid];
        float s = g;
        #pragma unroll
        for (int off = 16; off > 0; off >>= 1) s += __shfl_xor(s, off, 32);
        if (lane == 0) red[wave] = s;
        __syncthreads();

        // a <- (1-lr) a - lr*C*g ;  b <- b - lr*C*sum(g)
        aL[tid] = (1.0f - lr) * aL[tid] - lr * C_SVR * g;
        if (tid == 0) {
            float tot = 0.0f;
            #pragma unroll
            for (int w = 0; w < 16; ++w) tot += red[w];
            bL -= lr * C_SVR * tot;
        }
        __syncthreads();
    }

    // w[d] = sum_t a[t] * X[b, t, d]   (coalesced column reads)
    float acc = 0.0f;
    const float* Xb = X + (size_t)b * NT * ND;
    for (int t = 0; t < NT; ++t)
        acc += aL[t] * Xb[(size_t)t * ND + tid];
    out[b * ND + tid] = acc;
}

// ---------------------------------------------------------------------------
extern "C" void kernel_launch(void* const* d_in, const int* in_sizes, int n_in,
                              void* d_out, int out_size, void* d_ws, size_t ws_size,
                              hipStream_t stream) {
    const float* in = (const float*)d_in[0];              // [B, D, T] fp32
    float* Xws = (float*)d_ws;                            // 64 MB: X[B,T,D]
    float* Kws = Xws + (size_t)NB * NT * ND;              // 64 MB: K[B,T,T]
    float* out = (float*)d_out;                           // [B, D, 1]

    // allow >64KB dynamic LDS for k_svr (host-side, deterministic, capture-safe)
    (void)hipFuncSetAttribute(reinterpret_cast<const void*>(&k_svr),
                              hipFuncAttributeMaxDynamicSharedMemorySize,
                              (int)KL_BYTES);

    k_cummean<<<(NB * ND) / 256, 256, 0, stream>>>(in, Xws);
    k_norm<<<NB * NT, 128, 0, stream>>>(Xws);
    k_gram<<<(NB * 256) / 8, 256, 0, stream>>>(Xws, Kws);     // 8 waves/block
    k_svr<<<NB, 512, KL_BYTES, stream>>>(Kws, Xws, out);
}